// RobustNet_88021059764607
// MI455X (gfx1250) — compile-verified
//
#include <hip/hip_runtime.h>
#include <cstdint>

namespace {
constexpr int kB = 2048;
constexpr int kD = 256;
constexpr int kO = 512;
constexpr int TILE_R = 64;
constexpr int TILE_C = 64;
constexpr int THREADS = 256;  // 8 wave32
}

// Two d-steps folded into one v_max3_num_f32 with |src| modifiers:
// 2 subs (VOPD-pairable) + 1 max3 per 2 updates = 1.5 VALU/update.
#define CHEB_UPD(ACC, XA, WA, XB, WB)                                   \
  do {                                                                  \
    const float _da = (XA) - (WA);                                      \
    const float _db = (XB) - (WB);                                      \
    (ACC) = fmaxf((ACC), fmaxf(fabsf(_da), fabsf(_db)));                \
  } while (0)

__global__ __launch_bounds__(THREADS)
void cheb_tile_kernel(const float* __restrict__ X,
                      const float* __restrict__ W,
                      float* __restrict__ out) {
  // 256 x 64 f32 weight tile = 64 KB LDS, staged once; whole K-loop runs from LDS.
  __shared__ __align__(16) float sW[kD * TILE_C];

  const int c0  = blockIdx.x * TILE_C;
  const int r0  = blockIdx.y * TILE_R;
  const int tid = threadIdx.x;

  // ---- Stage the weight tile with gfx1250 async global->LDS b128 copies ----
  // Tile = 64 KB; 256 threads x 16 B = 4 KB per round -> 16 rounds.
  {
    const uint32_t lds_base = (uint32_t)(uintptr_t)(&sW[0]);  // low 32 bits of LDS flat addr = LDS offset
#pragma unroll
    for (int p = 0; p < (kD * TILE_C * 4) / (THREADS * 16); ++p) {
      const int idx = p * THREADS + tid;        // 16-byte granule index
      const int d   = idx >> 4;                 // tile row (64 floats = 16 granules)
      const int col = (idx & 15) << 2;          // float col within tile row
      const float* gp = W + (size_t)d * kO + (size_t)(c0 + col);
      const uint32_t lds_off = lds_base + (uint32_t)(idx << 4);
      const uint64_t ga = (uint64_t)(uintptr_t)gp;
      asm volatile("global_load_async_to_lds_b128 %0, %1, off"
                   :: "v"(lds_off), "v"(ga)
                   : "memory");
    }
    asm volatile("s_wait_asynccnt 0" ::: "memory");  // our copies have landed in LDS
  }
  __syncthreads();  // cross-wave visibility of the staged tile

  // Thread -> 4x4 register tile of outputs.
  const int tc = tid & 15;   // column group: lanes 0..15 of each wave
  const int tr = tid >> 4;   // row group: uniform across lanes 0..15 -> broadcast X loads
  const int cc = tc << 2;
  const float* xb = X + (size_t)(r0 + (tr << 2)) * kD;

  float acc[4][4];
#pragma unroll
  for (int r = 0; r < 4; ++r)
#pragma unroll
    for (int c = 0; c < 4; ++c) acc[r][c] = 0.0f;  // |.| >= 0, so 0 is the max identity

#pragma unroll 2
  for (int d = 0; d < kD; d += 4) {
    const float4 w0 = *reinterpret_cast<const float4*>(&sW[(d + 0) * TILE_C + cc]);
    const float4 w1 = *reinterpret_cast<const float4*>(&sW[(d + 1) * TILE_C + cc]);
    const float4 w2 = *reinterpret_cast<const float4*>(&sW[(d + 2) * TILE_C + cc]);
    const float4 w3 = *reinterpret_cast<const float4*>(&sW[(d + 3) * TILE_C + cc]);
#pragma unroll
    for (int r = 0; r < 4; ++r) {
      const float4 xv = *reinterpret_cast<const float4*>(xb + (size_t)r * kD + d);
      // 4 d-steps per accumulator = 4 subs + 2 max3.
      CHEB_UPD(acc[r][0], xv.x, w0.x, xv.y, w1.x);
      CHEB_UPD(acc[r][0], xv.z, w2.x, xv.w, w3.x);
      CHEB_UPD(acc[r][1], xv.x, w0.y, xv.y, w1.y);
      CHEB_UPD(acc[r][1], xv.z, w2.y, xv.w, w3.y);
      CHEB_UPD(acc[r][2], xv.x, w0.z, xv.y, w1.z);
      CHEB_UPD(acc[r][2], xv.z, w2.z, xv.w, w3.z);
      CHEB_UPD(acc[r][3], xv.x, w0.w, xv.y, w1.w);
      CHEB_UPD(acc[r][3], xv.z, w2.w, xv.w, w3.w);
    }
  }

  // Coalesced float4 stores: lanes 0..15 cover 256 contiguous bytes per row.
  float* ob = out + (size_t)(r0 + (tr << 2)) * kO + (size_t)(c0 + cc);
#pragma unroll
  for (int r = 0; r < 4; ++r) {
    float4 v;
    v.x = acc[r][0]; v.y = acc[r][1]; v.z = acc[r][2]; v.w = acc[r][3];
    *reinterpret_cast<float4*>(ob + (size_t)r * kO) = v;
  }
}

extern "C" void kernel_launch(void* const* d_in, const int* in_sizes, int n_in,
                              void* d_out, int out_size, void* d_ws, size_t ws_size,
                              hipStream_t stream) {
  (void)in_sizes; (void)n_in; (void)d_ws; (void)ws_size; (void)out_size;
  const float* X = (const float*)d_in[0];       // [B, D] f32
  const float* W = (const float*)d_in[1];       // [D, O] f32
  float* out = (float*)d_out;                   // [B, O] f32
  dim3 grid(kO / TILE_C, kB / TILE_R);          // (8, 32) = 256 blocks
  cheb_tile_kernel<<<grid, THREADS, 0, stream>>>(X, W, out);
}